// NET_7533372637184
// MI455X (gfx1250) — compile-verified
//
#include <hip/hip_runtime.h>
#include <math.h>

typedef __attribute__((ext_vector_type(2))) float v2f;
typedef __attribute__((ext_vector_type(8))) float v8f;

#define HEADS     8
#define IN_DIM    128
#define CH        32          // HID == OUT == 32
#define FDIM      256         // HEADS * CH
#define NEG_SLOPE 0.2f

// ---------------------------------------------------------------------------
// fp32 WMMA GEMM:  C[M x 256] = A[M x K] @ B[K x 256]
// One 128-thread block = 4 waves; wave w computes the 16x16 tile at
// (blockIdx.x*16, blockIdx.y*64 + w*16).  K-loop steps of 4 using
// V_WMMA_F32_16X16X4_F32 (full fp32 precision, matrix pipe).
//
// A-fragment layout (ISA 7.12.2, 32-bit A 16x4): lanes 0-15 = rows, VGPR0
// holds K=+0 (low half) / K=+2 (high half), VGPR1 holds K=+1 / K=+3 -> each
// lane loads a contiguous float2 at column k0 + 2*half.  B mirrored.
// C/D: VGPR r holds row (r + 8*half), column = lane&15.
// ---------------------------------------------------------------------------
__global__ void gemm_wmma_f32(const float* __restrict__ A,
                              const float* __restrict__ B,
                              float* __restrict__ C,
                              int M, int K) {
  const int lane = threadIdx.x & 31;
  const int wave = threadIdx.x >> 5;
  const int half = lane >> 4;
  const int l16  = lane & 15;
  const int tileM = blockIdx.x * 16;
  const int tileN = blockIdx.y * 64 + wave * 16;

  int arow = tileM + l16;
  if (arow >= M) arow = M - 1;          // clamp: WMMA needs full EXEC
  const int bcol = tileN + l16;

  v8f acc = {};
  for (int k0 = 0; k0 < K; k0 += 4) {
    v2f a, b;
    const float* ap = A + (size_t)arow * K + (k0 + 2 * half);
    a.x = ap[0];
    a.y = ap[1];
    const float* bp = B + (size_t)(k0 + 2 * half) * FDIM + bcol;
    b.x = bp[0];
    b.y = bp[FDIM];
    acc = __builtin_amdgcn_wmma_f32_16x16x4_f32(
        /*neg_a=*/false, a, /*neg_b=*/false, b,
        /*c_mod=*/(short)0, acc, /*reuse_a=*/false, /*reuse_b=*/false);
  }
#pragma unroll
  for (int r = 0; r < 8; ++r) {
    const int m = tileM + r + 8 * half;
    if (m < M) C[(size_t)m * FDIM + bcol] = acc[r];
  }
}

// ---------------------------------------------------------------------------
// Per-(node, head) attention coefficients: a = sum_c xh[n,h,c] * att[h,c]
// ---------------------------------------------------------------------------
__global__ void attn_coeff_kernel(const float* __restrict__ xh,
                                  const float* __restrict__ att_src,
                                  const float* __restrict__ att_dst,
                                  float* __restrict__ a_src,
                                  float* __restrict__ a_dst, int N) {
  const int idx = blockIdx.x * blockDim.x + threadIdx.x;
  if (idx >= N * HEADS) return;
  const int n = idx >> 3, h = idx & 7;
  const float4* row = (const float4*)(xh + (size_t)n * FDIM + h * CH);
  const float4* as  = (const float4*)(att_src + h * CH);
  const float4* ad  = (const float4*)(att_dst + h * CH);
  float s = 0.f, d = 0.f;
#pragma unroll
  for (int q = 0; q < CH / 4; ++q) {
    const float4 v = row[q], a = as[q], b = ad[q];
    s += v.x * a.x + v.y * a.y + v.z * a.z + v.w * a.w;
    d += v.x * b.x + v.y * b.y + v.z * b.z + v.w * b.w;
  }
  a_src[idx] = s;
  a_dst[idx] = d;
}

// ---------------------------------------------------------------------------
// Per-layer scratch init: agg = 0 (N*256), mmax = -inf, denom = 0 (N*8 each)
// ---------------------------------------------------------------------------
__global__ void init_kernel(float* __restrict__ agg, float* __restrict__ mmax,
                            float* __restrict__ denom, int N) {
  const int idx = blockIdx.x * blockDim.x + threadIdx.x;
  if (idx < N * FDIM) agg[idx] = 0.f;
  if (idx < N * HEADS) { mmax[idx] = -INFINITY; denom[idx] = 0.f; }
}

// IEEE-order-preserving float atomic max via int punning (always compiles).
__device__ __forceinline__ void atomicMaxF(float* addr, float v) {
  if (v >= 0.f) atomicMax((int*)addr, __float_as_int(v));
  else          atomicMin((unsigned int*)addr, __float_as_uint(v));
}

__device__ __forceinline__ void edge_sd(const int* __restrict__ ei, int e,
                                        int E, int& s, int& d) {
  if (e < E) { s = ei[e]; d = ei[E + e]; }
  else       { s = e - E; d = e - E; }          // self-loop
}

__device__ __forceinline__ float leaky(float v) {
  return v > 0.f ? v : NEG_SLOPE * v;
}

// Pass 1: segment max of leaky-relu logits over incoming edges
__global__ void edge_max_kernel(const int* __restrict__ ei,
                                const float* __restrict__ a_src,
                                const float* __restrict__ a_dst,
                                float* __restrict__ mmax, int E, int Et) {
  const int idx = blockIdx.x * blockDim.x + threadIdx.x;
  if (idx >= Et * HEADS) return;
  const int e = idx >> 3, h = idx & 7;
  int s, d; edge_sd(ei, e, E, s, d);
  const float v = leaky(a_src[s * HEADS + h] + a_dst[d * HEADS + h]);
  atomicMaxF(&mmax[d * HEADS + h], v);
}

// Pass 2: ee = exp(v - max); segment-sum denominators
__global__ void edge_expsum_kernel(const int* __restrict__ ei,
                                   const float* __restrict__ a_src,
                                   const float* __restrict__ a_dst,
                                   const float* __restrict__ mmax,
                                   float* __restrict__ ee,
                                   float* __restrict__ denom, int E, int Et) {
  const int idx = blockIdx.x * blockDim.x + threadIdx.x;
  if (idx >= Et * HEADS) return;
  const int e = idx >> 3, h = idx & 7;
  int s, d; edge_sd(ei, e, E, s, d);
  const float v = leaky(a_src[s * HEADS + h] + a_dst[d * HEADS + h]);
  const float x = __expf(v - mmax[d * HEADS + h]);
  ee[idx] = x;
  atomicAdd(&denom[d * HEADS + h], x);
}

// Pass 3: agg[dst,h,:] += (ee/denom) * xh[src,h,:]
__global__ void edge_agg_kernel(const int* __restrict__ ei,
                                const float* __restrict__ ee,
                                const float* __restrict__ denom,
                                const float* __restrict__ xh,
                                float* __restrict__ agg, int E, int Et) {
  const int idx = blockIdx.x * blockDim.x + threadIdx.x;
  if (idx >= Et * HEADS) return;
  const int e = idx >> 3, h = idx & 7;
  int s, d; edge_sd(ei, e, E, s, d);
  const float coeff = ee[idx] / denom[d * HEADS + h];
  const float4* srow = (const float4*)(xh + (size_t)s * FDIM + h * CH);
  float* drow = agg + (size_t)d * FDIM + h * CH;
#pragma unroll
  for (int q = 0; q < CH / 4; ++q) {
    const float4 v = srow[q];
    atomicAdd(&drow[4 * q + 0], coeff * v.x);
    atomicAdd(&drow[4 * q + 1], coeff * v.y);
    atomicAdd(&drow[4 * q + 2], coeff * v.z);
    atomicAdd(&drow[4 * q + 3], coeff * v.w);
  }
}

// Mean over heads + bias (+ optional ReLU)
__global__ void head_mean_kernel(const float* __restrict__ agg,
                                 const float* __restrict__ bias,
                                 float* __restrict__ out, int N, int do_relu) {
  const int idx = blockIdx.x * blockDim.x + threadIdx.x;
  if (idx >= N * CH) return;
  const int n = idx / CH, c = idx % CH;
  float acc = 0.f;
#pragma unroll
  for (int h = 0; h < HEADS; ++h) acc += agg[(size_t)n * FDIM + h * CH + c];
  float o = acc * (1.0f / HEADS) + bias[c];
  if (do_relu) o = fmaxf(o, 0.f);
  out[idx] = o;
}

// Decode: per candidate pair 32-dim dot product
__global__ void decode_kernel(const int* __restrict__ eli,
                              const float* __restrict__ z,
                              float* __restrict__ out, int EL) {
  const int p = blockIdx.x * blockDim.x + threadIdx.x;
  if (p >= EL) return;
  const int i = eli[p], j = eli[EL + p];
  const float4* zi = (const float4*)(z + (size_t)i * CH);
  const float4* zj = (const float4*)(z + (size_t)j * CH);
  float s = 0.f;
#pragma unroll
  for (int q = 0; q < CH / 4; ++q) {
    const float4 a = zi[q], b = zj[q];
    s += a.x * b.x + a.y * b.y + a.z * b.z + a.w * b.w;
  }
  out[p] = s;
}

// ---------------------------------------------------------------------------
// Host-side layer driver (launches only; graph-capture safe)
// ---------------------------------------------------------------------------
static void run_gat_layer(const float* in, int K, const float* W,
                          const float* att_src, const float* att_dst,
                          const float* bias, float* out_nodes, int do_relu,
                          const int* ei, int N, int E,
                          float* xh, float* agg, float* a_src, float* a_dst,
                          float* mmax, float* denom, float* ee,
                          hipStream_t stream) {
  const int Et = E + N;
  // GEMM: xh = in @ W   (M=N rows, K, 256 cols)
  dim3 ggrid((N + 15) / 16, FDIM / 64);
  gemm_wmma_f32<<<ggrid, 128, 0, stream>>>(in, W, xh, N, K);
  // attention coefficients
  int t = N * HEADS;
  attn_coeff_kernel<<<(t + 255) / 256, 256, 0, stream>>>(xh, att_src, att_dst,
                                                         a_src, a_dst, N);
  // init scratch
  init_kernel<<<(N * FDIM + 255) / 256, 256, 0, stream>>>(agg, mmax, denom, N);
  // edge passes
  t = Et * HEADS;
  edge_max_kernel<<<(t + 255) / 256, 256, 0, stream>>>(ei, a_src, a_dst, mmax,
                                                       E, Et);
  edge_expsum_kernel<<<(t + 255) / 256, 256, 0, stream>>>(ei, a_src, a_dst,
                                                          mmax, ee, denom, E, Et);
  edge_agg_kernel<<<(t + 255) / 256, 256, 0, stream>>>(ei, ee, denom, xh, agg,
                                                       E, Et);
  // head mean + bias (+relu)
  t = N * CH;
  head_mean_kernel<<<(t + 255) / 256, 256, 0, stream>>>(agg, bias, out_nodes,
                                                        N, do_relu);
}

extern "C" void kernel_launch(void* const* d_in, const int* in_sizes, int n_in,
                              void* d_out, int out_size, void* d_ws,
                              size_t ws_size, hipStream_t stream) {
  (void)n_in; (void)ws_size;
  const float* x    = (const float*)d_in[0];
  const int*   ei   = (const int*)d_in[1];
  const int*   eli  = (const int*)d_in[2];
  const float* W1   = (const float*)d_in[3];
  const float* as1  = (const float*)d_in[4];
  const float* ad1  = (const float*)d_in[5];
  const float* b1   = (const float*)d_in[6];
  const float* W2   = (const float*)d_in[7];
  const float* as2  = (const float*)d_in[8];
  const float* ad2  = (const float*)d_in[9];
  const float* b2   = (const float*)d_in[10];
  float* out = (float*)d_out;

  const int N  = in_sizes[0] / IN_DIM;
  const int E  = in_sizes[1] / 2;
  const int EL = in_sizes[2] / 2;
  const int Et = E + N;

  // workspace carve-up (floats)
  float* ws    = (float*)d_ws;
  float* xh    = ws;                         // N*256
  float* agg   = xh    + (size_t)N * FDIM;   // N*256
  float* a_src = agg   + (size_t)N * FDIM;   // N*8
  float* a_dst = a_src + (size_t)N * HEADS;  // N*8
  float* mmax  = a_dst + (size_t)N * HEADS;  // N*8
  float* denom = mmax  + (size_t)N * HEADS;  // N*8
  float* ee    = denom + (size_t)N * HEADS;  // Et*8
  float* hbuf  = ee    + (size_t)Et * HEADS; // N*32
  float* zbuf  = hbuf  + (size_t)N * CH;     // N*32

  // layer 1: GATConv(128 -> 32, heads=8, mean) + ReLU
  run_gat_layer(x, IN_DIM, W1, as1, ad1, b1, hbuf, /*relu=*/1, ei, N, E,
                xh, agg, a_src, a_dst, mmax, denom, ee, stream);
  // layer 2: GATConv(32 -> 32, heads=8, mean)
  run_gat_layer(hbuf, CH, W2, as2, ad2, b2, zbuf, /*relu=*/0, ei, N, E,
                xh, agg, a_src, a_dst, mmax, denom, ee, stream);
  // decode
  decode_kernel<<<(EL + 255) / 256, 256, 0, stream>>>(eli, zbuf, out, EL);
  (void)out_size;
}